// DifferentiableJPEG_25812753449603
// MI455X (gfx1250) — compile-verified
//
#include <hip/hip_runtime.h>

typedef __attribute__((ext_vector_type(2))) float v2f;
typedef __attribute__((ext_vector_type(8))) float v8f;

// 8x8 DCT basis (row-major)
__constant__ float cDCT[64] = {
  0.3536f, 0.3536f, 0.3536f, 0.3536f, 0.3536f, 0.3536f, 0.3536f, 0.3536f,
  0.4904f, 0.4157f, 0.2778f, 0.0975f,-0.0975f,-0.2778f,-0.4157f,-0.4904f,
  0.4619f, 0.1913f,-0.1913f,-0.4619f,-0.4619f,-0.1913f, 0.1913f, 0.4619f,
  0.4157f,-0.0975f,-0.4904f,-0.2778f, 0.2778f, 0.4904f, 0.0975f,-0.4157f,
  0.3536f,-0.3536f,-0.3536f, 0.3536f, 0.3536f,-0.3536f,-0.3536f, 0.3536f,
  0.2778f,-0.4904f, 0.0975f, 0.4157f,-0.4157f,-0.0975f, 0.4904f,-0.2778f,
  0.1913f,-0.4619f, 0.4619f,-0.1913f,-0.1913f, 0.4619f,-0.4619f, 0.1913f,
  0.0975f,-0.2778f, 0.4157f,-0.4904f, 0.4904f,-0.4157f, 0.2778f,-0.0975f
};

// F2 = blockdiag(F, F), 16x16
static __device__ __forceinline__ float f2elem(int a, int b) {
  return ((a ^ b) & 8) ? 0.0f : cDCT[(a & 7) * 8 + (b & 7)];
}

// D = sum_k A[k](16x4) * B[k](4x16) over K=16, f32 WMMA
static __device__ __forceinline__ v8f mm16(const v2f A[4], const v2f B[4]) {
  v8f acc = {0.f, 0.f, 0.f, 0.f, 0.f, 0.f, 0.f, 0.f};
#pragma unroll
  for (int k = 0; k < 4; ++k)
    acc = __builtin_amdgcn_wmma_f32_16x16x4_f32(false, A[k], false, B[k],
                                                (short)0, acc, false, false);
  return acc;
}

// Convert a 16x16 matrix from C/D layout (v8f) to B layout (4 x v2f).
// Row kk lives at {slice kk>>2, vgpr kk&1, lane-half (kk>>1)&1}; column index
// stays in lane[3:0], so only half<->half shuffles are needed.
static __device__ __forceinline__ void c_to_b(v8f m, int hi, int p15, v2f out[4]) {
#pragma unroll
  for (int k = 0; k < 4; ++k) {
#pragma unroll
    for (int j = 0; j < 2; ++j) {
      const int r0 = 4 * k + j;      // row needed by lane-half 0
      const int r1 = 4 * k + j + 2;  // row needed by lane-half 1
      float v0 = __shfl(m[r0 & 7], ((r0 >> 3) << 4) | p15, 32);
      float v1 = __shfl(m[r1 & 7], ((r1 >> 3) << 4) | p15, 32);
      out[k][j] = hi ? v1 : v0;
    }
  }
}

__global__ __launch_bounds__(256)
void jpeg_fused_kernel(const float* __restrict__ x,
                       const float* __restrict__ qlum,
                       const float* __restrict__ qchr,
                       float* __restrict__ outY,
                       float* __restrict__ outXq) {
  __shared__ float lds[8][16 * 17];  // per-wave transpose tile, stride 17 (conflict-free)

  const int tid  = threadIdx.x;
  const int w    = tid >> 5;
  const int lane = tid & 31;
  const int hi   = lane >> 4;   // lane half
  const int p15  = lane & 15;   // A-layout row / C-layout column

  const int gw   = blockIdx.x * 8 + w;    // 0..32767
  const int b    = gw >> 10;              // batch
  const int tile = gw & 1023;
  const int ty   = tile >> 5, tx = tile & 31;
  const int y0   = ty << 4, x0 = tx << 4;

  // Constant fragments: Creg = F2 (as A) == F2^T (as B); Treg = F2^T (as A) == F2 (as B)
  v2f Creg[4], Treg[4];
#pragma unroll
  for (int k = 0; k < 4; ++k)
#pragma unroll
    for (int j = 0; j < 2; ++j) {
      const int c = 4 * k + 2 * hi + j;
      Creg[k][j] = f2elem(p15, c);
      Treg[k][j] = f2elem(c, p15);
    }

  // Quant tables: column (lane&7), all 8 rows; q index for C-layout elem is [vgpr][lane&7]
  float qv[2][8], qinv[2][8];
#pragma unroll
  for (int r = 0; r < 8; ++r) {
    qv[0][r] = qlum[r * 8 + (lane & 7)];
    qv[1][r] = qchr[r * 8 + (lane & 7)];
    qinv[0][r] = 1.0f / qv[0][r];
    qinv[1][r] = 1.0f / qv[1][r];
  }

  // Load 16x16 RGB tile directly in A layout; fuse RGB->YCbCr (-127 centering)
  v2f aP[3][4];
  const float* row = x + (((b * 512) + y0 + p15) * 512 + x0) * 3;
#pragma unroll
  for (int k = 0; k < 4; ++k) {
    const int c0 = 4 * k + 2 * hi;
#pragma unroll
    for (int j = 0; j < 2; ++j) {
      const float* px = row + (c0 + j) * 3;
      float R = px[0] * 255.0f, G = px[1] * 255.0f, B = px[2] * 255.0f;
      aP[0][k][j] =  0.299f    * R + 0.587f    * G + 0.114f    * B - 127.0f;
      aP[1][k][j] = -0.168736f * R - 0.331264f * G + 0.5f      * B + 1.0f;
      aP[2][k][j] =  0.5f      * R - 0.418688f * G - 0.081312f * B + 1.0f;
    }
  }

  float* ldsw = lds[w];
  float Rrec[3][8];

#pragma unroll
  for (int ch = 0; ch < 3; ++ch) {
    const int qs = ch ? 1 : 0;

    // ---- DCT: X = F2 @ (P @ F2^T) ----
    v8f m1 = mm16(aP[ch], Creg);          // P (A) x F2^T (B)
    v2f m1b[4];
    c_to_b(m1, hi, p15, m1b);
    v8f X = mm16(Creg, m1b);              // F2 (A) x M1 (B)  -> C layout

    // ---- quantize (sine-series round), emit Xq, stage for transpose ----
    // C-layout element at vgpr r: (m = r + 8*hi, n = p15)
    const int nblk  = (2 * ty + hi) * 64 + 2 * tx + ((lane >> 3) & 1);
    const int xbase = (((b * 3 + ch) * 4096) + nblk) * 64 + (lane & 7);
#pragma unroll
    for (int r = 0; r < 8; ++r) {
      float t = X[r] * qinv[qs][r];
      float rs = t
        - 0.31830988618f * __sinf( 6.28318530718f * t)
        + 0.15915494309f * __sinf(12.56637061436f * t)
        - 0.10610329539f * __sinf(18.84955592154f * t)
        + 0.07957747155f * __sinf(25.13274122872f * t)
        - 0.06366197724f * __sinf(31.41592653590f * t);
      float xq = rs * qv[qs][r];
      outXq[xbase + r * 8] = xq;
      ldsw[(r + 8 * hi) * 17 + p15] = xq;   // full (m,n) matrix into LDS
    }

    // ---- read Xq back in A layout (true transpose via LDS) ----
    v2f aXq[4];
#pragma unroll
    for (int k = 0; k < 4; ++k)
#pragma unroll
      for (int j = 0; j < 2; ++j)
        aXq[k][j] = ldsw[p15 * 17 + 4 * k + 2 * hi + j];

    // ---- IDCT: R = F2^T @ (Xq @ F2) ----
    v8f m2 = mm16(aXq, Treg);             // Xq (A) x F2 (B)
    v2f m2b[4];
    c_to_b(m2, hi, p15, m2b);
    v8f Rv = mm16(Treg, m2b);             // F2^T (A) x M2 (B)
#pragma unroll
    for (int r = 0; r < 8; ++r) Rrec[ch][r] = Rv[r];
  }

  // ---- YCbCr -> RGB, clip, store y ----
#pragma unroll
  for (int r = 0; r < 8; ++r) {
    float Y  = Rrec[0][r] + 127.0f;
    float Cb = Rrec[1][r] + 127.0f;
    float Cr = Rrec[2][r] + 127.0f;
    float rr = (-179.456f    + Y + 1.402f * Cr) * (1.0f / 255.0f);
    float gg = ( 135.458816f + Y - 0.344136f * Cb - 0.714136f * Cr) * (1.0f / 255.0f);
    float bb = (-226.816f    + Y + 1.772f * Cb) * (1.0f / 255.0f);
    rr = fminf(fmaxf(rr, 0.0f), 1.0f);
    gg = fminf(fmaxf(gg, 0.0f), 1.0f);
    bb = fminf(fmaxf(bb, 0.0f), 1.0f);
    const int m = r + 8 * hi;
    float* o = outY + (((b * 512) + y0 + m) * 512 + x0 + p15) * 3;
    o[0] = rr; o[1] = gg; o[2] = bb;
  }
}

extern "C" void kernel_launch(void* const* d_in, const int* in_sizes, int n_in,
                              void* d_out, int out_size, void* d_ws, size_t ws_size,
                              hipStream_t stream) {
  (void)in_sizes; (void)n_in; (void)out_size; (void)d_ws; (void)ws_size;
  const float* x   = (const float*)d_in[0];
  const float* ql  = (const float*)d_in[1];
  const float* qc  = (const float*)d_in[2];
  float* outY  = (float*)d_out;
  float* outXq = outY + (size_t)32 * 512 * 512 * 3;  // y first, then Xq
  // 32768 tile-waves = 4096 blocks x 8 waves (256 threads)
  jpeg_fused_kernel<<<dim3(4096), dim3(256), 0, stream>>>(x, ql, qc, outY, outXq);
}